// PointCloudRenderer_8753143349744
// MI455X (gfx1250) — compile-verified
//
#include <hip/hip_runtime.h>
#include <hip/hip_bf16.h>

// Point-cloud splat renderer for MI455X (gfx1250, wave32).
//   - LDS-resident scatter-max via ds_max_num_f32 (the workload's bottleneck path)
//   - 8 row-bands per (batch,view) image -> 768 blocks x 256 threads for occupancy
//   - constant trig tables (angles are compile-time fixed) -> no scalar libm bloat
//   - write-out via GLOBAL_STORE_ASYNC_FROM_LDS_B128 (ASYNCcnt) when available

#define IMG      224
#define NPTS     32768
#define NBATCH   16
#define NVIEWS   6
#define BANDS    8
#define ROWS     (IMG / BANDS)      // 28
#define BAND_PIX (ROWS * IMG)       // 6272 floats = 25088 B LDS
#define BLOCK    256

#if defined(__AMDGCN__) && __has_builtin(__builtin_amdgcn_global_store_async_from_lds_b128)
  #define ASYNC_STORE_OK 1
#else
  #define ASYNC_STORE_OK 0
#endif

__global__ __launch_bounds__(BLOCK, 4)
void pc_render_kernel(const float* __restrict__ pts, float* __restrict__ out) {
    __shared__ __align__(16) float img[BAND_PIX];
    __shared__ float red[2];   // [0] = zmin, [1] = zmax

    const int tid  = threadIdx.x;
    const int band = blockIdx.x % BANDS;
    const int bv   = blockIdx.x / BANDS;
    const int v    = bv % NVIEWS;
    const int b    = bv / NVIEWS;

    // Compile-time trig: az = v*60deg, el = {0,30,-30,0,0,0} deg.
    // cos/sin(0,60,120,180,240,300 deg) and cos/sin(+-30 deg).
    const float S60 = 0.86602540378443865f;
    const float ca = (v == 0) ? 1.0f  : (v == 3) ? -1.0f
                   : (v == 1 || v == 5) ? 0.5f : -0.5f;
    const float sa = (v == 0 || v == 3) ? 0.0f
                   : (v == 1 || v == 2) ? S60 : -S60;
    const float ce = (v == 1 || v == 2) ? S60 : 1.0f;
    const float se = (v == 1) ? 0.5f : (v == 2) ? -0.5f : 0.0f;

    // Init band image (reference scatters into zeros; feat >= 0.3 > 0)
    for (int i = tid; i < BAND_PIX; i += BLOCK) img[i] = 0.0f;
    if (tid == 0) { red[0] = 3.4e38f; red[1] = -3.4e38f; }
    __syncthreads();

    const float* __restrict__ P = pts + (size_t)b * NPTS * 3;

    // ---- Pass A: zmin / zmax of rotated depth over the whole batch ----
    float zmn = 3.4e38f, zmx = -3.4e38f;
    for (int p = tid; p < NPTS; p += BLOCK) {
        const float x = P[3 * p + 0];
        const float y = P[3 * p + 1];
        const float z = P[3 * p + 2];
        const float zr = x * sa + z * ca;
        const float zf = y * se + zr * ce;
        zmn = fminf(zmn, zf);
        zmx = fmaxf(zmx, zf);
    }
    // wave32 butterfly reduce, then one LDS float atomic per wave
    #pragma unroll
    for (int o = 16; o > 0; o >>= 1) {
        zmn = fminf(zmn, __shfl_xor(zmn, o));
        zmx = fmaxf(zmx, __shfl_xor(zmx, o));
    }
    if ((tid & 31) == 0) {
        __hip_atomic_fetch_min(&red[0], zmn, __ATOMIC_RELAXED, __HIP_MEMORY_SCOPE_WORKGROUP);
        __hip_atomic_fetch_max(&red[1], zmx, __ATOMIC_RELAXED, __HIP_MEMORY_SCOPE_WORKGROUP);
    }
    __syncthreads();

    const float zmin = red[0];
    const float inv  = 0.7f / (red[1] - zmin + 1e-6f);

    // ---- Pass B: rotate + splat into this block's band ----
    const int r0 = band * ROWS;
    const int r1 = r0 + ROWS;
    for (int p = tid; p < NPTS; p += BLOCK) {
        __builtin_prefetch(&P[3 * (p + BLOCK)], 0, 1);  // global_prefetch_b8
        const float x = P[3 * p + 0];
        const float y = P[3 * p + 1];
        const float z = P[3 * p + 2];
        const float xr = x * ca - z * sa;
        const float zr = x * sa + z * ca;
        const float yr = y * ce - zr * se;
        const float zf = y * se + zr * ce;
        const float feat = 0.3f + (zf - zmin) * inv;

        // 5 candidate px / py values (separable 5x5 splat). Offsets span only
        // +-1 pixel, truncation is monotone -> dedupe vs previous candidate.
        int pxs[5], pys[5];
        #pragma unroll
        for (int i = 0; i < 5; i++) {
            const float dd = (float)(i - 2) * (1.0f / 224.0f);
            pxs[i] = (int)((xr + dd + 1.0f) * 0.5f * 223.0f);
            pys[i] = (int)((yr + dd + 1.0f) * 0.5f * 223.0f);
        }
        #pragma unroll
        for (int iy = 0; iy < 5; iy++) {
            const int py = pys[iy];
            if (iy && py == pys[iy - 1]) continue;   // duplicate row
            if (py < r0 || py >= r1) continue;       // outside this band (also OOB cull)
            const int rowoff = (py - r0) * IMG;
            #pragma unroll
            for (int ix = 0; ix < 5; ix++) {
                const int px = pxs[ix];
                if (ix && px == pxs[ix - 1]) continue;  // duplicate col
                if ((unsigned)px >= (unsigned)IMG) continue;
                // ds_max_num_f32 on LDS
                __hip_atomic_fetch_max(&img[rowoff + px], feat,
                                       __ATOMIC_RELAXED, __HIP_MEMORY_SCOPE_WORKGROUP);
            }
        }
    }
    __syncthreads();

    // ---- Write band out, replicated to 3 channels. Layout (B,V,3,H,W). ----
    float* __restrict__ ob =
        out + ((size_t)(b * NVIEWS + v) * 3) * (IMG * IMG) + (size_t)r0 * IMG;

#if ASYNC_STORE_OK
    // Builtin signature (from clang diagnostic): arg0 = int4 addrspace(1)*
    // (printed as "__device__"), arg1 = int4 addrspace(3)*, then imm offset,
    // imm cpol. Cast through uintptr_t to the exact vector pointer types.
    typedef int v4i __attribute__((vector_size(16)));
    typedef __attribute__((address_space(1))) v4i* gv4p;   // global int4*
    typedef __attribute__((address_space(3))) v4i* lv4p;   // LDS int4*
    // 16 B per lane per op; all offsets are 16-B aligned.
    const int nvec = BAND_PIX / 4;                         // 1568 float4 chunks
    for (int i = tid; i < nvec; i += BLOCK) {
        const float* l = &img[4 * i];
        float*       g = ob + 4 * i;
        __builtin_amdgcn_global_store_async_from_lds_b128(
            (gv4p)(uintptr_t)g, (lv4p)(uintptr_t)l, 0, 0);
        __builtin_amdgcn_global_store_async_from_lds_b128(
            (gv4p)(uintptr_t)(g + IMG * IMG), (lv4p)(uintptr_t)l, 0, 0);
        __builtin_amdgcn_global_store_async_from_lds_b128(
            (gv4p)(uintptr_t)(g + 2 * IMG * IMG), (lv4p)(uintptr_t)l, 0, 0);
    }
  #if __has_builtin(__builtin_amdgcn_s_wait_asynccnt)
    __builtin_amdgcn_s_wait_asynccnt(0);
  #endif
#else
    for (int i = tid; i < BAND_PIX; i += BLOCK) {
        const float f = img[i];
        ob[i] = f;
        ob[IMG * IMG + i] = f;
        ob[2 * IMG * IMG + i] = f;
    }
#endif
}

extern "C" void kernel_launch(void* const* d_in, const int* in_sizes, int n_in,
                              void* d_out, int out_size, void* d_ws, size_t ws_size,
                              hipStream_t stream) {
    (void)in_sizes; (void)n_in; (void)out_size; (void)d_ws; (void)ws_size;
    const float* pts = (const float*)d_in[0];   // (16, 32768, 3) f32
    float* out       = (float*)d_out;           // (16, 6, 3, 224, 224) f32

    const int grid = NBATCH * NVIEWS * BANDS;   // 768 blocks
    pc_render_kernel<<<dim3(grid), dim3(BLOCK), 0, stream>>>(pts, out);
}